// CodeEmbeddingModule_65214783422482
// MI455X (gfx1250) — compile-verified
//
#include <hip/hip_runtime.h>
#include <stdint.h>

#define N_SEQ 256
#define LSEQ  512
#define DDIM  128
#define EDIM  128
#define CORE_TERM 50000

// Native clang vectors: legal for nontemporal builtins and async-LDS builtins.
typedef float v4f __attribute__((ext_vector_type(4)));
typedef int   v4i __attribute__((vector_size(4 * sizeof(int))));

// ---- CDNA5 async global<->LDS DMA availability probe (device pass only) ----
#if defined(__AMDGCN__) && \
    __has_builtin(__builtin_amdgcn_global_load_async_to_lds_b128) && \
    __has_builtin(__builtin_amdgcn_global_store_async_from_lds_b128)
#define USE_ASYNC_LDS 1
#else
#define USE_ASYNC_LDS 0
#endif

#if USE_ASYNC_LDS
typedef __attribute__((address_space(1))) v4i v4i_glb;  // printed as __device__
typedef __attribute__((address_space(3))) v4i v4i_lds;  // printed as __shared__

__device__ __forceinline__ void async_wait0() {
#if __has_builtin(__builtin_amdgcn_s_wait_asynccnt)
    __builtin_amdgcn_s_wait_asynccnt(0);
#else
    asm volatile("s_wait_asynccnt 0" ::: "memory");
#endif
}
#endif

// ---------------------------------------------------------------------------
// Kernel A (defined first so the disasm snippet shows it): fused permuted
// concat + embedding gather/add.
//   out[rank[i], l, 0:128]   = mat[i, l, :]                (async LDS DMA path)
//   out[rank[i], l, 128:256] = emb[ct[i,l]] + emb[50000+l] (VALU add, NT store)
// Grid: (LSEQ/8, N_SEQ). Block: 256 threads = 8 waves; wave w owns l = l0 + w,
// lane q owns float4 q of the 128-float row.
// ---------------------------------------------------------------------------
__global__ __launch_bounds__(256) void ced_gather_concat_kernel(
    const float* __restrict__ mat,   // [N_SEQ, LSEQ, DDIM]
    const int*   __restrict__ ct,    // [N_SEQ, LSEQ]
    const float* __restrict__ emb,   // [CORE_TERM + LSEQ, EDIM]
    const int*   __restrict__ rank,  // [N_SEQ]
    float* __restrict__ out)         // [N_SEQ, LSEQ, DDIM+EDIM]
{
    const int i  = blockIdx.y;
    const int l0 = blockIdx.x * 8;
    const int t  = threadIdx.x;
    const int ll = t >> 5;         // wave index within block: 0..7
    const int q  = t & 31;         // float4 index within a 128-float row
    const int l  = l0 + ll;
    const int r  = rank[i];

    const v4f* mat4 = (const v4f*)mat;
    const v4f* emb4 = (const v4f*)emb;
    v4f*       out4 = (v4f*)out;

    const int mat_idx = (i * LSEQ + l) * (DDIM / 4) + q;
    const int out_row = (r * LSEQ + l) * ((DDIM + EDIM) / 4);

#if USE_ASYNC_LDS
    // ---- matrix half: global -> LDS -> global via async DMA (no VGPR data) --
    __shared__ v4f tile[256];                // 16 B per lane, per-lane private
    v4i_lds* lp = (v4i_lds*)&tile[t];
    v4i_glb* gs = (v4i_glb*)(mat4 + mat_idx);
    v4i_glb* gd = (v4i_glb*)(out4 + out_row + q);
    __builtin_amdgcn_global_load_async_to_lds_b128(gs, lp, 0, 0);
    async_wait0();                           // LDS slot filled for this wave
    __builtin_amdgcn_global_store_async_from_lds_b128(gd, lp, 0, 0);
    // s_endpgm performs an implicit wait-idle, draining the async store.
#else
    out4[out_row + q] = mat4[mat_idx];
#endif

    // ---- embedding half: gather + positional add, non-temporal store -------
    const int token = ct[i * LSEQ + l];      // uniform per wave
    const v4f e = emb4[token * (EDIM / 4) + q];
    const v4f p = emb4[(CORE_TERM + l) * (EDIM / 4) + q];
    const v4f s = e + p;
    __builtin_nontemporal_store(s, out4 + out_row + (DDIM / 4) + q);
}

// ---------------------------------------------------------------------------
// Kernel B: stable descending rank of 256 lengths.
//   rank[i] == idx_unsort[i]; length_sorted[rank[i]] = length[i]
// ---------------------------------------------------------------------------
__global__ __launch_bounds__(N_SEQ) void ced_rank_kernel(
    const int* __restrict__ length,
    int* __restrict__ rank,             // workspace, N_SEQ ints
    float* __restrict__ out_len_sorted, // output tail: length_sorted as float
    float* __restrict__ out_idx_unsort) // output tail: idx_unsort as float
{
    __shared__ int s_len[N_SEQ];
    const int i = threadIdx.x;
    s_len[i] = length[i];
    __syncthreads();
    const int li = s_len[i];
    int r = 0;
#pragma unroll 8
    for (int j = 0; j < N_SEQ; ++j) {
        const int lj = s_len[j];
        r += (lj > li) || ((lj == li) && (j < i));
    }
    rank[i] = r;
    out_len_sorted[r] = (float)li;
    out_idx_unsort[i] = (float)r;
}

// ---------------------------------------------------------------------------
extern "C" void kernel_launch(void* const* d_in, const int* in_sizes, int n_in,
                              void* d_out, int out_size, void* d_ws, size_t ws_size,
                              hipStream_t stream) {
    const float* mat    = (const float*)d_in[0];  // [16,16,512,128] f32
    const int*   length = (const int*)  d_in[1];  // [256] i32
    const int*   ct     = (const int*)  d_in[2];  // [16,16,512] i32
    const float* emb    = (const float*)d_in[3];  // [50512,128] f32

    float* out = (float*)d_out;
    // d_out layout: x [256*512*256] | length_sorted [256] | idx_unsort [256]
    float* out_len_sorted = out + (size_t)N_SEQ * LSEQ * (DDIM + EDIM);
    float* out_idx_unsort = out_len_sorted + N_SEQ;

    int* rank = (int*)d_ws;  // N_SEQ ints of scratch

    ced_rank_kernel<<<1, N_SEQ, 0, stream>>>(length, rank, out_len_sorted,
                                             out_idx_unsort);

    dim3 grid(LSEQ / 8, N_SEQ);
    ced_gather_concat_kernel<<<grid, 256, 0, stream>>>(mat, ct, emb, rank, out);
}